// FraudRGCN_25941602467851
// MI455X (gfx1250) — compile-verified
//
#include <hip/hip_runtime.h>

#define NNODES 100000
#define NEDGES 600000
#define FDIM   128
#define NREL   3
#define LDSS   516   // 512 + 4 pad floats: row stride not multiple of 64 banks

typedef float v2f __attribute__((ext_vector_type(2)));
typedef float v8f __attribute__((ext_vector_type(8)));

// ---------------- zero scratch (float4 grid-stride) ----------------
__global__ void zero_f4(float* __restrict__ p, long n4) {
  long i = (long)blockIdx.x * blockDim.x + threadIdx.x;
  long stride = (long)gridDim.x * blockDim.x;
  float4 z = make_float4(0.f, 0.f, 0.f, 0.f);
  for (; i < n4; i += stride) ((float4*)p)[i] = z;
}

// ---------------- per-relation edge scatter (mean numerator + counts) ----
// one wave per edge; lane l handles feature cols [4l, 4l+4)
__global__ void edge_agg(const int* __restrict__ ei, const int* __restrict__ et,
                         const float* __restrict__ feat, float* __restrict__ agg,
                         float* __restrict__ cnt, int do_cnt) {
  int lane = threadIdx.x & 31;
  int wave = (int)((blockIdx.x * blockDim.x + threadIdx.x) >> 5);
  int nwaves = (int)((gridDim.x * blockDim.x) >> 5);
  for (int e = wave; e < NEDGES; e += nwaves) {
    int s = ei[e];
    int d = ei[NEDGES + e];
    int r = et[e];
    float4 v = *(const float4*)(feat + (long)s * FDIM + lane * 4);
    float* a = agg + ((long)r * NNODES + d) * FDIM + lane * 4;
    atomicAdd(a + 0, v.x);
    atomicAdd(a + 1, v.y);
    atomicAdd(a + 2, v.z);
    atomicAdd(a + 3, v.w);
    if (do_cnt && lane == 0) atomicAdd(cnt + (long)r * NNODES + d, 1.0f);
  }
}

// ---------------- divide by max(count,1) ----------------
__global__ void normalize_agg(float* __restrict__ agg, const float* __restrict__ cnt) {
  long g = (long)blockIdx.x * blockDim.x + threadIdx.x;  // one float4 each
  long row = g >> 5;                                     // 32 float4 per 128-col row
  int c4 = (int)(g & 31);
  if (row >= (long)NREL * NNODES) return;
  float s = 1.0f / fmaxf(cnt[row], 1.0f);
  float4* p = (float4*)(agg + row * FDIM) + c4;
  float4 v = *p;
  v.x *= s; v.y *= s; v.z *= s; v.w *= s;
  *p = v;
}

// ---------------- fused RGCN layer GEMM: [A0|A1|A2|A3](N x 512) @ [B0;B1;B2;B3](512 x 128)
// + bias, ReLU.  One 16x16 f32 WMMA C-tile per wave, 8 waves -> 16 x 128 per block.
__global__ __launch_bounds__(256) void rgcn_gemm(
    const float* __restrict__ A0, const float* __restrict__ A1,
    const float* __restrict__ A2, const float* __restrict__ A3,
    const float* __restrict__ B0, const float* __restrict__ B1,
    const float* __restrict__ B2, const float* __restrict__ B3,
    const float* __restrict__ bias, float* __restrict__ out) {
  __shared__ __align__(16) float Alds[16 * LDSS];
  int t = threadIdx.x;
  long row0 = (long)blockIdx.x * 16;

  // stage 16 x 512 A-tile (4 segments of 128) into LDS; coalesced float4 loads
  #pragma unroll
  for (int i = 0; i < 8; ++i) {
    int f4  = t + 256 * i;   // 0..2047, 128 float4 per LDS row
    int row = f4 >> 7;
    int c4  = f4 & 127;
    int seg = c4 >> 5;
    int cw  = c4 & 31;
    const float* sp = (seg == 0) ? A0 : (seg == 1) ? A1 : (seg == 2) ? A2 : A3;
    float4 v = *(const float4*)(sp + (row0 + row) * FDIM + cw * 4);
    *(float4*)(&Alds[row * LDSS + c4 * 4]) = v;
  }
  __syncthreads();

  int wave = t >> 5;
  int lane = t & 31;
  int half = lane >> 4;        // K sub-block select per ISA f32 A/B layout
  int mn   = lane & 15;
  int col0 = wave * 16;

  v8f acc = {};
  const float* Bs[4] = {B0, B1, B2, B3};
  #pragma unroll
  for (int seg = 0; seg < 4; ++seg) {
    const float* B  = Bs[seg];
    const float* Ab = &Alds[mn * LDSS + seg * 128 + 2 * half];
    #pragma unroll 4
    for (int kk = 0; kk < 128; kk += 4) {
      v2f a = *(const v2f*)(Ab + kk);            // A[m][k0+2h], A[m][k0+2h+1]
      int kr = kk + 2 * half;
      v2f b;
      b.x = B[kr * FDIM + col0 + mn];            // B[k0+2h][n]
      b.y = B[(kr + 1) * FDIM + col0 + mn];      // B[k0+2h+1][n]
      acc = __builtin_amdgcn_wmma_f32_16x16x4_f32(
          /*neg_a=*/false, a, /*neg_b=*/false, b,
          /*c_mod=*/(short)0, acc, /*reuse_a=*/false, /*reuse_b=*/false);
    }
  }

  // epilogue: bias + ReLU, C layout: VGPR i -> M = i + 8*half, N = mn
  int col = col0 + mn;
  float bc = bias[col];
  #pragma unroll
  for (int i = 0; i < 8; ++i) {
    int m = i + 8 * half;
    out[(row0 + m) * FDIM + col] = fmaxf(acc[i] + bc, 0.0f);
  }
}

// ---------------- head: logits = h @ lin_w + lin_b, log_softmax (C=2) ------
__global__ void head_logsoftmax(const float* __restrict__ h, const float* __restrict__ lw,
                                const float* __restrict__ lb, float* __restrict__ out) {
  int lane = threadIdx.x & 31;
  int node = (int)((blockIdx.x * blockDim.x + threadIdx.x) >> 5);
  if (node >= NNODES) return;
  float4 v = *(const float4*)(h + (long)node * FDIM + lane * 4);
  const float* w = lw + lane * 8;   // (k,c) interleaved, k = 4*lane..4*lane+3
  float p0 = v.x * w[0] + v.y * w[2] + v.z * w[4] + v.w * w[6];
  float p1 = v.x * w[1] + v.y * w[3] + v.z * w[5] + v.w * w[7];
  #pragma unroll
  for (int off = 16; off > 0; off >>= 1) {
    p0 += __shfl_xor(p0, off, 32);
    p1 += __shfl_xor(p1, off, 32);
  }
  if (lane == 0) {
    float l0 = p0 + lb[0], l1 = p1 + lb[1];
    float m = fmaxf(l0, l1);
    float lse = m + logf(expf(l0 - m) + expf(l1 - m));
    out[(long)node * 2 + 0] = l0 - lse;
    out[(long)node * 2 + 1] = l1 - lse;
  }
}

extern "C" void kernel_launch(void* const* d_in, const int* in_sizes, int n_in,
                              void* d_out, int out_size, void* d_ws, size_t ws_size,
                              hipStream_t stream) {
  const float* x      = (const float*)d_in[0];
  const int*   ei     = (const int*)d_in[1];
  const int*   et     = (const int*)d_in[2];
  const float* W1rel  = (const float*)d_in[3];
  const float* W1root = (const float*)d_in[4];
  const float* b1     = (const float*)d_in[5];
  const float* W2rel  = (const float*)d_in[6];
  const float* W2root = (const float*)d_in[7];
  const float* b2     = (const float*)d_in[8];
  const float* lw     = (const float*)d_in[9];
  const float* lb     = (const float*)d_in[10];

  const long aggN = (long)NREL * NNODES * FDIM;  // 38.4M floats
  const long cntN = (long)NREL * NNODES;         // 0.3M floats
  const long NF   = (long)NNODES * FDIM;         // 12.8M floats

  float* ws  = (float*)d_ws;
  float* agg = ws;              // per-relation mean accumulators
  float* cnt = agg + aggN;      // per-relation degree counts
  float* h1  = cnt + cntN;      // layer-1 activations
  float* h2  = h1 + NF;         // layer-2 activations

  dim3 blk(256);

  // ---- layer 1 ----
  zero_f4<<<4096, blk, 0, stream>>>(agg, (aggN + cntN) / 4);  // agg + cnt contiguous
  edge_agg<<<2048, blk, 0, stream>>>(ei, et, x, agg, cnt, /*do_cnt=*/1);
  normalize_agg<<<(int)(cntN * 32 / 256), blk, 0, stream>>>(agg, cnt);
  rgcn_gemm<<<NNODES / 16, blk, 0, stream>>>(
      x, agg, agg + NF, agg + 2 * NF,
      W1root, W1rel, W1rel + FDIM * FDIM, W1rel + 2 * FDIM * FDIM, b1, h1);

  // ---- layer 2 (counts reused; re-zero accumulators only) ----
  zero_f4<<<4096, blk, 0, stream>>>(agg, aggN / 4);
  edge_agg<<<2048, blk, 0, stream>>>(ei, et, h1, agg, cnt, /*do_cnt=*/0);
  normalize_agg<<<(int)(cntN * 32 / 256), blk, 0, stream>>>(agg, cnt);
  rgcn_gemm<<<NNODES / 16, blk, 0, stream>>>(
      h1, agg, agg + NF, agg + 2 * NF,
      W2root, W2rel, W2rel + FDIM * FDIM, W2rel + 2 * FDIM * FDIM, b2, h2);

  // ---- head ----
  head_logsoftmax<<<(NNODES * 32) / 256, blk, 0, stream>>>(h2, lw, lb, (float*)d_out);
}